// MockOlmoeTopKRouter_25022479466896
// MI455X (gfx1250) — compile-verified
//
#include <hip/hip_runtime.h>
#include <hip/hip_bf16.h>
#include <math.h>

typedef __attribute__((ext_vector_type(2))) float v2f;
typedef __attribute__((ext_vector_type(8))) float v8f;

#define TOKENS 16384
#define HID    2048
#define NEXP   64
#define TOPK   8

// ---------------------------------------------------------------------------
// GEMM: logits[t][e] = sum_k H[t][k] * W[e][k]
// Block = 256 threads = 8 waves. Waves: 2 M-tiles x 4 N-tiles.
// Each wave owns a 16(token) x 16(expert) tile, full K = 2048 via
// V_WMMA_F32_16X16X4_F32 (512 steps of K=4).
//
// Fragment layout (f32, 16x16x4, wave32):
//   A (16x4, 2 VGPRs): lane<16: M=lane, K={0,1}; lane>=16: M=lane-16, K={2,3}
//   B (4x16, 2 VGPRs): lane<16: N=lane, K={0,1}; lane>=16: N=lane-16, K={2,3}
//   -> both operands: 2 consecutive floats at row (base + lane&15),
//      col (k + 2*(lane>>4)), since H and W are both row-major with K contiguous.
//   C/D (16x16, 8 VGPRs): vgpr j: lane<16 -> (M=j, N=lane); lane>=16 -> (M=j+8, N=lane-16)
// ---------------------------------------------------------------------------
__global__ __launch_bounds__(256) void router_gemm_kernel(
    const float* __restrict__ H, const float* __restrict__ W,
    float* __restrict__ logits)
{
    const int wave = threadIdx.x >> 5;     // 0..7
    const int lane = threadIdx.x & 31;
    const int r    = lane & 15;
    const int h    = lane >> 4;            // 0 or 1

    const int mtile = blockIdx.x * 2 + (wave >> 2);
    const int etile = wave & 3;
    const int m0 = mtile * 16;
    const int e0 = etile * 16;

    const float* aptr = H + (size_t)(m0 + r) * HID + 2 * h;
    const float* bptr = W + (size_t)(e0 + r) * HID + 2 * h;

    v8f acc = {};

#pragma unroll 8
    for (int k = 0; k < HID; k += 4) {
        v2f a = *(const v2f*)aptr;
        v2f b = *(const v2f*)bptr;
        // 8 args: (neg_a, A, neg_b, B, c_mod, C, reuse_a, reuse_b)
        acc = __builtin_amdgcn_wmma_f32_16x16x4_f32(
            false, a, false, b, (short)0, acc, false, false);
        aptr += 4;
        bptr += 4;
    }

    // Write D: acc[j] belongs to token (m0 + j + 8*h), expert (e0 + r).
#pragma unroll
    for (int j = 0; j < 8; ++j) {
        logits[(size_t)(m0 + j + 8 * h) * NEXP + (e0 + r)] = acc[j];
    }
}

// ---------------------------------------------------------------------------
// Per-token top-8 (sorted descending, lowest index wins ties) + softmax.
// One thread per token; 64 logits held in registers.
// ---------------------------------------------------------------------------
__global__ __launch_bounds__(256) void topk_softmax_kernel(
    const float* __restrict__ logits,
    float* __restrict__ weights_out,
    float* __restrict__ experts_out)
{
    const int t = blockIdx.x * blockDim.x + threadIdx.x;
    if (t >= TOKENS) return;

    float v[NEXP];
    const float4* src = (const float4*)(logits + (size_t)t * NEXP);
#pragma unroll
    for (int i = 0; i < NEXP / 4; ++i) {
        float4 x = src[i];
        v[4 * i + 0] = x.x;
        v[4 * i + 1] = x.y;
        v[4 * i + 2] = x.z;
        v[4 * i + 3] = x.w;
    }

    float topv[TOPK];
    int   topi[TOPK];
    unsigned long long used = 0ull;

    for (int j = 0; j < TOPK; ++j) {
        float best = -INFINITY;
        int   bi   = 0;
#pragma unroll
        for (int i = 0; i < NEXP; ++i) {
            // strict > keeps the lowest index on ties (torch/jax top_k order)
            bool ok = !((used >> i) & 1ull) && (v[i] > best);
            if (ok) { best = v[i]; bi = i; }
        }
        used |= (1ull << bi);
        topv[j] = best;
        topi[j] = bi;
    }

    // softmax over the 8 selected logits; topv[0] is the max.
    const float m = topv[0];
    float e[TOPK];
    float sum = 0.0f;
#pragma unroll
    for (int j = 0; j < TOPK; ++j) {
        e[j] = __expf(topv[j] - m);
        sum += e[j];
    }
    const float inv = 1.0f / sum;
#pragma unroll
    for (int j = 0; j < TOPK; ++j) {
        weights_out[(size_t)t * TOPK + j] = e[j] * inv;
        experts_out[(size_t)t * TOPK + j] = (float)topi[j];
    }
}

extern "C" void kernel_launch(void* const* d_in, const int* in_sizes, int n_in,
                              void* d_out, int out_size, void* d_ws, size_t ws_size,
                              hipStream_t stream) {
    const float* H = (const float*)d_in[0];   // [16384, 2048]
    const float* W = (const float*)d_in[1];   // [64, 2048]

    float* out = (float*)d_out;
    float* rw  = out;                               // routing_weights [16384, 8]
    float* se  = out + (size_t)TOKENS * TOPK;       // selected_experts [16384, 8]
    float* lg  = out + (size_t)2 * TOKENS * TOPK;   // router_logits [16384, 64]

    // 2 M-tiles (32 tokens) per block of 256 threads.
    router_gemm_kernel<<<TOKENS / 32, 256, 0, stream>>>(H, W, lg);
    topk_softmax_kernel<<<TOKENS / 256, 256, 0, stream>>>(lg, rw, se);
}